// LetterGNN_3908420240264
// MI455X (gfx1250) — compile-verified
//
#include <hip/hip_runtime.h>
#include <hip/hip_bf16.h>

#define N_NODES   100000
#define N_EDGES   1600000
#define FEAT      128
#define N_GRAPHS  64
#define N_CLASSES 32

typedef float v2f __attribute__((ext_vector_type(2)));
typedef float v8f __attribute__((ext_vector_type(8)));
typedef __attribute__((address_space(3))) void lds_void;

// Native f32 atomic add, no return, device scope (RMW resolves in L2).
__device__ __forceinline__ void atomAddF(float* p, float v) {
    asm volatile("global_atomic_add_f32 %0, %1, off scope:SCOPE_DEV"
                 :: "v"(p), "v"(v) : "memory");
}

// CDNA5 async copy: 16B global -> LDS, tracked by ASYNCcnt (no VGPR data path).
__device__ __forceinline__ void async_copy_b128(float* lds, const float* g) {
    unsigned off = (unsigned)(uintptr_t)(lds_void*)lds;   // LDS byte offset (AS3)
    asm volatile("global_load_async_to_lds_b128 %0, %1, off"
                 :: "v"(off), "v"(g) : "memory");
}

__device__ __forceinline__ void wait_async_all() {
    asm volatile("s_wait_asynccnt 0x0" ::: "memory");
}

// ---------------------------------------------------------------------------
// Degree: deg[dst] += 1 over all edges (deg buffer pre-zeroed)
// ---------------------------------------------------------------------------
__global__ void deg_kernel(const int* __restrict__ dst, float* __restrict__ deg, int nedges) {
    int i = blockIdx.x * blockDim.x + threadIdx.x;
    if (i < nedges) atomAddF(&deg[dst[i]], 1.0f);
}

// dinv = rsqrt(deg + 1)   (in place)
__global__ void dinv_kernel(float* __restrict__ deg, int n) {
    int i = blockIdx.x * blockDim.x + threadIdx.x;
    if (i < n) deg[i] = rsqrtf(deg[i] + 1.0f);
}

// ---------------------------------------------------------------------------
// H = X @ W   (X: [n x 128] row-major, W: [128 x 128] row-major)
// Block: 256 threads = 8 waves. Block tile: 64 rows x 128 cols.
// Wave w owns columns [16w, 16w+16); 4 row-tiles of 16 -> 4 v8f accumulators.
// Tiles staged via GLOBAL_LOAD_ASYNC_TO_LDS_B128; V_WMMA_F32_16X16X4_F32 core.
// ---------------------------------------------------------------------------
__global__ __launch_bounds__(256) void gemm_xw(const float* __restrict__ X,
                                               const float* __restrict__ W,
                                               float* __restrict__ H,
                                               int nrows) {
    __shared__ float Xs[64 * 36];   // 64 rows x 32 k, stride 36 (16B aligned, bank-spread)
    __shared__ float Ws[32 * 136];  // 32 k   x 128 cols, stride 136

    const int tid  = threadIdx.x;
    const int lane = tid & 31;
    const int wave = tid >> 5;                 // 0..7 -> column tile
    const int rowbase = blockIdx.x * 64;
    const bool full = (rowbase + 64 <= nrows); // uniform: all 64 rows valid

    v8f acc[4] = {};                           // zero-init accumulators

    const int ml   = lane & 15;                // M index within tile (A/C) / col (B)
    const int koff = (lane >> 4) << 1;         // 0 or 2 (K sub-offset per A/B layout)
    const int col  = (wave << 4) + ml;         // column within 128-wide block tile

    for (int kb = 0; kb < 4; ++kb) {
        // ---- stage X tile: 64 x 32 ----
        if (full) {
            #pragma unroll
            for (int i = 0; i < 2; ++i) {
                int idx = tid + i * 256;       // 0..511
                int r   = idx >> 3;            // 0..63
                int p   = idx & 7;             // 0..7 (float4 chunk)
                async_copy_b128(Xs + r * 36 + p * 4,
                                X + (size_t)(rowbase + r) * FEAT + kb * 32 + p * 4);
            }
        } else {
            #pragma unroll
            for (int i = 0; i < 2; ++i) {
                int idx  = tid + i * 256;
                int r    = idx >> 3;
                int p    = idx & 7;
                int grow = rowbase + r;
                float4 v = make_float4(0.f, 0.f, 0.f, 0.f);
                if (grow < nrows)
                    v = *(const float4*)(X + (size_t)grow * FEAT + kb * 32 + p * 4);
                *(float4*)(Xs + r * 36 + p * 4) = v;
            }
        }
        // ---- stage W tile: 32 x 128 (always in range) ----
        #pragma unroll
        for (int i = 0; i < 4; ++i) {
            int idx = tid + i * 256;           // 0..1023
            int r   = idx >> 5;                // 0..31
            int c   = idx & 31;                // 0..31 (float4 chunk)
            async_copy_b128(Ws + r * 136 + c * 4,
                            W + (size_t)(kb * 32 + r) * FEAT + c * 4);
        }
        // prefetch next X tile (global_prefetch_b8)
        if (kb < 3) {
            int r = tid >> 3, p = tid & 7;
            int grow = rowbase + r;
            if (grow < nrows)
                __builtin_prefetch(X + (size_t)grow * FEAT + (kb + 1) * 32 + p * 4, 0, 0);
        }
        wait_async_all();
        __syncthreads();

        // ---- compute: 8 k-steps x 4 row tiles of V_WMMA_F32_16X16X4_F32 ----
        #pragma unroll
        for (int k0 = 0; k0 < 32; k0 += 4) {
            // B fragment 4x16: v0 = rows {k0 | k0+2}, v1 = rows {k0+1 | k0+3}
            v2f bf;
            bf.x = Ws[(k0 + koff) * 136 + col];
            bf.y = Ws[(k0 + koff + 1) * 136 + col];
            #pragma unroll
            for (int r = 0; r < 4; ++r) {
                // A fragment 16x4: lane m holds K = k0+koff, k0+koff+1
                const float2 a2 = *(const float2*)(Xs + ((r << 4) + ml) * 36 + k0 + koff);
                v2f af; af.x = a2.x; af.y = a2.y;
                acc[r] = __builtin_amdgcn_wmma_f32_16x16x4_f32(
                    false, af, false, bf, (short)0, acc[r], false, false);
            }
        }
        __syncthreads();
    }

    // ---- store: C/D layout VGPR v -> rows {v | v+8}, lanes 0..15 / 16..31 ----
    const int mh = (lane >> 4) << 3;           // 0 or 8
    #pragma unroll
    for (int r = 0; r < 4; ++r) {
        int base = rowbase + (r << 4) + mh;
        #pragma unroll
        for (int v = 0; v < 8; ++v) {
            int grow = base + v;
            if (grow < nrows) H[(size_t)grow * FEAT + col] = acc[r][v];
        }
    }
}

// ---------------------------------------------------------------------------
// Edge scatter: agg[dst] += h[src] * dinv[src]   (one wave per edge)
// ---------------------------------------------------------------------------
__global__ __launch_bounds__(256) void edge_scatter(const float* __restrict__ h,
                                                    const int* __restrict__ src,
                                                    const int* __restrict__ dst,
                                                    const float* __restrict__ dinv,
                                                    float* __restrict__ agg,
                                                    int nedges) {
    const int lane = threadIdx.x & 31;
    int warp = (blockIdx.x * blockDim.x + threadIdx.x) >> 5;
    const int nwarps = (gridDim.x * blockDim.x) >> 5;
    for (int e = warp; e < nedges; e += nwarps) {
        int s = src[e];                        // uniform across wave -> broadcast loads
        int d = dst[e];
        float w = dinv[s];
        float4 v = *(const float4*)(h + (size_t)s * FEAT + lane * 4);
        float* p = agg + (size_t)d * FEAT + lane * 4;
        atomAddF(p + 0, v.x * w);
        atomAddF(p + 1, v.y * w);
        atomAddF(p + 2, v.z * w);
        atomAddF(p + 3, v.w * w);
    }
}

// ---------------------------------------------------------------------------
// Epilogue: z = relu(dinv*agg + dinv^2*h + b)   (one wave per node)
// ---------------------------------------------------------------------------
__global__ __launch_bounds__(256) void post_kernel(const float* __restrict__ h,
                                                   const float* __restrict__ agg,
                                                   const float* __restrict__ dinv,
                                                   const float* __restrict__ b,
                                                   float* __restrict__ z, int n) {
    const int lane = threadIdx.x & 31;
    int node = (blockIdx.x * blockDim.x + threadIdx.x) >> 5;
    if (node >= n) return;
    float di  = dinv[node];
    float di2 = di * di;
    float4 a  = *(const float4*)(agg + (size_t)node * FEAT + lane * 4);
    float4 hv = *(const float4*)(h + (size_t)node * FEAT + lane * 4);
    float4 bv = *(const float4*)(b + lane * 4);
    float4 o;
    o.x = fmaxf(di * a.x + di2 * hv.x + bv.x, 0.f);
    o.y = fmaxf(di * a.y + di2 * hv.y + bv.y, 0.f);
    o.z = fmaxf(di * a.z + di2 * hv.z + bv.z, 0.f);
    o.w = fmaxf(di * a.w + di2 * hv.w + bv.w, 0.f);
    *(float4*)(z + (size_t)node * FEAT + lane * 4) = o;
}

// ---------------------------------------------------------------------------
// Pooling: pooled[g] += z[node]; cnt[g] += 1    (one wave per node)
// ---------------------------------------------------------------------------
__global__ __launch_bounds__(256) void pool_kernel(const float* __restrict__ z,
                                                   const int* __restrict__ batch,
                                                   float* __restrict__ pooled,
                                                   float* __restrict__ cnt, int n) {
    const int lane = threadIdx.x & 31;
    int node = (blockIdx.x * blockDim.x + threadIdx.x) >> 5;
    if (node >= n) return;
    int g = batch[node];
    float4 v = *(const float4*)(z + (size_t)node * FEAT + lane * 4);
    float* p = pooled + (size_t)g * FEAT + lane * 4;
    atomAddF(p + 0, v.x);
    atomAddF(p + 1, v.y);
    atomAddF(p + 2, v.z);
    atomAddF(p + 3, v.w);
    if (lane == 0) atomAddF(cnt + g, 1.0f);
}

// ---------------------------------------------------------------------------
// FC head: out[g][c] = (pooled[g]/max(cnt,1)) @ fcW + fcb   (tiny: 64x32)
// ---------------------------------------------------------------------------
__global__ void fc_kernel(const float* __restrict__ pooled, const float* __restrict__ cnt,
                          const float* __restrict__ fcW, const float* __restrict__ fcb,
                          float* __restrict__ out) {
    int idx = blockIdx.x * blockDim.x + threadIdx.x;
    if (idx >= N_GRAPHS * N_CLASSES) return;
    int g = idx >> 5;          // N_CLASSES == 32
    int c = idx & 31;
    float inv = 1.0f / fmaxf(cnt[g], 1.0f);
    float acc = 0.0f;
    #pragma unroll 8
    for (int k = 0; k < FEAT; ++k)
        acc += pooled[(size_t)g * FEAT + k] * fcW[(size_t)k * N_CLASSES + c];
    out[idx] = acc * inv + fcb[c];
}

// ---------------------------------------------------------------------------
extern "C" void kernel_launch(void* const* d_in, const int* in_sizes, int n_in,
                              void* d_out, int out_size, void* d_ws, size_t ws_size,
                              hipStream_t stream) {
    const float* X   = (const float*)d_in[0];
    const int*   edg = (const int*)d_in[1];
    const int*   bat = (const int*)d_in[2];
    const float* W1  = (const float*)d_in[3];
    const float* b1  = (const float*)d_in[4];
    const float* W2  = (const float*)d_in[5];
    const float* b2  = (const float*)d_in[6];
    const float* fcW = (const float*)d_in[7];
    const float* fcb = (const float*)d_in[8];
    float* out = (float*)d_out;

    const int* src = edg;
    const int* dst = edg + N_EDGES;

    const size_t NF = (size_t)N_NODES * FEAT;
    float* h      = (float*)d_ws;
    float* agg    = h + NF;
    float* z      = agg + NF;
    float* dinv   = z + NF;                 // doubles as deg
    float* pooled = dinv + N_NODES;
    float* cnt    = pooled + N_GRAPHS * FEAT;

    // zero scratch that is accumulated into (graph-capture legal)
    hipMemsetAsync(dinv, 0, N_NODES * sizeof(float), stream);
    hipMemsetAsync(agg, 0, NF * sizeof(float), stream);
    hipMemsetAsync(pooled, 0, (N_GRAPHS * FEAT + N_GRAPHS) * sizeof(float), stream);

    const int nodeWarpBlocks = (N_NODES * 32 + 255) / 256;   // one wave per node
    const int gemmBlocks     = (N_NODES + 63) / 64;

    // degrees -> dinv
    deg_kernel<<<(N_EDGES + 255) / 256, 256, 0, stream>>>(dst, dinv, N_EDGES);
    dinv_kernel<<<(N_NODES + 255) / 256, 256, 0, stream>>>(dinv, N_NODES);

    // ---- layer 1 ----
    gemm_xw<<<gemmBlocks, 256, 0, stream>>>(X, W1, h, N_NODES);
    edge_scatter<<<16384, 256, 0, stream>>>(h, src, dst, dinv, agg, N_EDGES);
    post_kernel<<<nodeWarpBlocks, 256, 0, stream>>>(h, agg, dinv, b1, z, N_NODES);

    // ---- layer 2 (reuse buffers) ----
    hipMemsetAsync(agg, 0, NF * sizeof(float), stream);
    gemm_xw<<<gemmBlocks, 256, 0, stream>>>(z, W2, h, N_NODES);
    edge_scatter<<<16384, 256, 0, stream>>>(h, src, dst, dinv, agg, N_EDGES);
    post_kernel<<<nodeWarpBlocks, 256, 0, stream>>>(h, agg, dinv, b2, z, N_NODES);

    // ---- pooling + FC head ----
    pool_kernel<<<nodeWarpBlocks, 256, 0, stream>>>(z, bat, pooled, cnt, N_NODES);
    fc_kernel<<<(N_GRAPHS * N_CLASSES + 255) / 256, 256, 0, stream>>>(pooled, cnt, fcW, fcb, out);
}